// GCN_6923487281238
// MI455X (gfx1250) — compile-verified
//
#include <hip/hip_runtime.h>
#include <hip/hip_bf16.h>
#include <stdint.h>

// Tile of edges staged into LDS per double-buffer slot.
constexpr int TILE = 1024;   // edges per tile
constexpr int TPB  = 256;    // threads per block (8 waves on wave32)

// ---------------------------------------------------------------------------
// Kernel 1: per-node MLP  h = relu(x*W1 + b1) @ W2 + b2, plus zero sum/cnt.
// ---------------------------------------------------------------------------
__global__ void gcn_init_mlp(const float* __restrict__ x,
                             const float* __restrict__ W1,
                             const float* __restrict__ b1,
                             const float* __restrict__ W2,
                             const float* __restrict__ b2,
                             float* __restrict__ h,
                             float* __restrict__ sum,
                             float* __restrict__ cnt,
                             int n) {
    const float w10 = W1[0], w11 = W1[1], w12 = W1[2], w13 = W1[3];
    const float a0  = b1[0], a1  = b1[1], a2  = b1[2], a3  = b1[3];
    const float v0  = W2[0], v1  = W2[1], v2  = W2[2], v3  = W2[3];
    const float c   = b2[0];
    for (int i = blockIdx.x * blockDim.x + threadIdx.x; i < n;
         i += gridDim.x * blockDim.x) {
        // x is read exactly once -> non-temporal, don't pollute L2.
        const float xv = __builtin_nontemporal_load(&x[i]);
        float acc = c;
        acc = fmaf(fmaxf(fmaf(xv, w10, a0), 0.0f), v0, acc);
        acc = fmaf(fmaxf(fmaf(xv, w11, a1), 0.0f), v1, acc);
        acc = fmaf(fmaxf(fmaf(xv, w12, a2), 0.0f), v2, acc);
        acc = fmaf(fmaxf(fmaf(xv, w13, a3), 0.0f), v3, acc);
        h[i]   = acc;    // hot: gathered by edge phase, keep cacheable
        sum[i] = 0.0f;   // hot: atomic targets, keep cacheable
        cnt[i] = 0.0f;
    }
}

// ---------------------------------------------------------------------------
// Kernel 2: edge scatter with async global->LDS double buffering.
// Each thread owns 4 edges of a 1024-edge tile. Each wave issues one
// global_load_async_to_lds_b128 per row (src / dst) per tile: 32 lanes x 16B
// = 512B = 128 edges per wave, 8 waves cover the tile. The 256MB edge stream
// is marked non-temporal so the 12MB h/sum/cnt hot set stays L2-resident.
// ---------------------------------------------------------------------------
__device__ __forceinline__ uint32_t lds_off(const void* p) {
    // Generic shared pointer: low 32 bits are the LDS byte offset.
    return (uint32_t)(uintptr_t)p;
}

__global__ void gcn_edge_scatter(const int* __restrict__ src,
                                 const int* __restrict__ dst,
                                 const float* __restrict__ h,
                                 float* __restrict__ sum,
                                 float* __restrict__ cnt,
                                 int n_tiles) {
    __shared__ __align__(16) int s_src[2][TILE];
    __shared__ __align__(16) int s_dst[2][TILE];

    const int tid = threadIdx.x;
    int tile = blockIdx.x;
    if (tile >= n_tiles) return;
    const int stride = gridDim.x;

    const uint32_t lane_b = (uint32_t)tid * 16u;   // 16B per lane
    const uint32_t lsrc0 = lds_off(&s_src[0][0]) + lane_b;
    const uint32_t lsrc1 = lds_off(&s_src[1][0]) + lane_b;
    const uint32_t ldst0 = lds_off(&s_dst[0][0]) + lane_b;
    const uint32_t ldst1 = lds_off(&s_dst[1][0]) + lane_b;

    // Prologue: issue first tile into buffer 0 (non-temporal stream).
    {
        const uint32_t voff = (uint32_t)tile * (TILE * 4u) + lane_b;
        asm volatile("global_load_async_to_lds_b128 %0, %1, %2 th:TH_LOAD_NT"
                     :: "v"(lsrc0), "v"(voff), "s"(src) : "memory");
        asm volatile("global_load_async_to_lds_b128 %0, %1, %2 th:TH_LOAD_NT"
                     :: "v"(ldst0), "v"(voff), "s"(dst) : "memory");
    }

    int buf = 0;
    for (; tile < n_tiles; tile += stride, buf ^= 1) {
        const int next = tile + stride;
        if (next < n_tiles) {
            // Issue next tile into the other buffer, then wait until only
            // those 2 async ops remain (async ops retire in order per wave).
            const uint32_t voff = (uint32_t)next * (TILE * 4u) + lane_b;
            const uint32_t ls = buf ? lsrc0 : lsrc1;
            const uint32_t ld = buf ? ldst0 : ldst1;
            asm volatile("global_load_async_to_lds_b128 %0, %1, %2 th:TH_LOAD_NT"
                         :: "v"(ls), "v"(voff), "s"(src) : "memory");
            asm volatile("global_load_async_to_lds_b128 %0, %1, %2 th:TH_LOAD_NT"
                         :: "v"(ld), "v"(voff), "s"(dst) : "memory");
            asm volatile("s_wait_asynccnt 2" ::: "memory");
        } else {
            asm volatile("s_wait_asynccnt 0" ::: "memory");
        }
        __syncthreads();   // all waves' slices of this tile are now in LDS

        const int4 sv = *(const int4*)&s_src[buf][tid * 4];
        const int4 dv = *(const int4*)&s_dst[buf][tid * 4];

        // Gathers hit L2 (h is 4MB << 192MB L2, and stays resident since the
        // edge stream is NT).
        const float h0 = h[sv.x];
        const float h1 = h[sv.y];
        const float h2 = h[sv.z];
        const float h3 = h[sv.w];

        // Scatter-add into L2-resident sum/cnt (no-return f32 atomics).
        atomicAdd(&sum[dv.x], h0);
        atomicAdd(&sum[dv.y], h1);
        atomicAdd(&sum[dv.z], h2);
        atomicAdd(&sum[dv.w], h3);
        atomicAdd(&cnt[dv.x], 1.0f);
        atomicAdd(&cnt[dv.y], 1.0f);
        atomicAdd(&cnt[dv.z], 1.0f);
        atomicAdd(&cnt[dv.w], 1.0f);

        __syncthreads();   // reads done before buffer is re-issued next iter
    }
}

// Remainder edges (E not a multiple of TILE) — plain path, NT edge reads.
__global__ void gcn_edge_rem(const int* __restrict__ src,
                             const int* __restrict__ dst,
                             const float* __restrict__ h,
                             float* __restrict__ sum,
                             float* __restrict__ cnt,
                             int start, int n_edges) {
    for (int e = start + blockIdx.x * blockDim.x + threadIdx.x; e < n_edges;
         e += gridDim.x * blockDim.x) {
        const int s = __builtin_nontemporal_load(&src[e]);
        const int d = __builtin_nontemporal_load(&dst[e]);
        atomicAdd(&sum[d], h[s]);
        atomicAdd(&cnt[d], 1.0f);
    }
}

// ---------------------------------------------------------------------------
// Kernel 3: out = (sum / max(cnt,1)) * Wsage
// ---------------------------------------------------------------------------
__global__ void gcn_finalize(const float* __restrict__ sum,
                             const float* __restrict__ cnt,
                             const float* __restrict__ wsage,
                             float* __restrict__ out,
                             int n) {
    const float w = wsage[0];
    for (int i = blockIdx.x * blockDim.x + threadIdx.x; i < n;
         i += gridDim.x * blockDim.x) {
        const float m = sum[i] / fmaxf(cnt[i], 1.0f);
        __builtin_nontemporal_store(m * w, &out[i]);
    }
}

extern "C" void kernel_launch(void* const* d_in, const int* in_sizes, int n_in,
                              void* d_out, int out_size, void* d_ws, size_t ws_size,
                              hipStream_t stream) {
    const float* x          = (const float*)d_in[0];
    const int*   edge_index = (const int*)d_in[1];
    const float* W1         = (const float*)d_in[2];
    const float* b1         = (const float*)d_in[3];
    const float* W2         = (const float*)d_in[4];
    const float* b2         = (const float*)d_in[5];
    const float* Wsage      = (const float*)d_in[6];
    float* out = (float*)d_out;

    const int n = in_sizes[0];        // x is [N,1]
    const int E = in_sizes[1] / 2;    // edge_index is [2,E]
    const int* src = edge_index;
    const int* dst = edge_index + E;

    // Workspace layout: h | sum | cnt  (3 * n floats = 12 MB)
    float* h   = (float*)d_ws;
    float* sum = h + n;
    float* cnt = sum + n;

    const int init_blocks = (n + TPB - 1) / TPB;
    gcn_init_mlp<<<init_blocks, TPB, 0, stream>>>(x, W1, b1, W2, b2,
                                                  h, sum, cnt, n);

    const int n_tiles = E / TILE;
    if (n_tiles > 0) {
        const int g = n_tiles < 4096 ? n_tiles : 4096;
        gcn_edge_scatter<<<g, TPB, 0, stream>>>(src, dst, h, sum, cnt, n_tiles);
    }
    const int rem_start = n_tiles * TILE;
    if (rem_start < E) {
        gcn_edge_rem<<<256, TPB, 0, stream>>>(src, dst, h, sum, cnt,
                                              rem_start, E);
    }

    gcn_finalize<<<init_blocks, TPB, 0, stream>>>(sum, cnt, Wsage, out, n);
}